// KernelActivation_64647847739998
// MI455X (gfx1250) — compile-verified
//
#include <hip/hip_runtime.h>
#include <stdint.h>

typedef float    v4f   __attribute__((ext_vector_type(4)));
typedef uint32_t u32x4 __attribute__((ext_vector_type(4)));
typedef uint32_t u32x8 __attribute__((ext_vector_type(8)));

namespace {
constexpr int kW               = 256;                         // image width in floats
constexpr int kChunkRows       = 64;                          // rows staged per workgroup
constexpr int kChunkFloats     = kW * kChunkRows;             // 16384 floats = 64 KB
constexpr int kPatchesPerChunk = (kChunkRows / 4) * (kW / 4); // 1024 patches / chunk
}

__device__ __forceinline__ v4f vmax4(v4f a, v4f b) {
  v4f r;
  r.x = fmaxf(a.x, b.x); r.y = fmaxf(a.y, b.y);
  r.z = fmaxf(a.z, b.z); r.w = fmaxf(a.w, b.w);
  return r;
}
__device__ __forceinline__ v4f vexp4(v4f a, float m) {
  v4f r;
  r.x = __expf(a.x - m); r.y = __expf(a.y - m);
  r.z = __expf(a.z - m); r.w = __expf(a.w - m);
  return r;
}
__device__ __forceinline__ float hsum4(v4f a) { return (a.x + a.y) + (a.z + a.w); }
__device__ __forceinline__ float hmax4(v4f a) { return fmaxf(fmaxf(a.x, a.y), fmaxf(a.z, a.w)); }

__global__ __launch_bounds__(256)
void patch_softmax_tdm(const float* __restrict__ x, float* __restrict__ out) {
  __shared__ float tile[kChunkFloats];  // 64 KB staging buffer

  const uint32_t chunk = blockIdx.x;

  // 32-bit LDS byte offset of `tile` (low 32 bits of the generic address are
  // the LDS offset). Feeding this into the asm also *captures* `tile`, so the
  // "memory" clobber is understood to write it — without this the compiler
  // proved the asm couldn't touch `tile` and DCE'd the entire softmax.
  const uint32_t lds_base = (uint32_t)(uintptr_t)tile;

  // ---- Wave 0: issue one TDM bulk copy (global -> LDS) for the whole chunk ----
  if (threadIdx.x == 0) {
    const uint64_t gaddr =
        (uint64_t)(uintptr_t)x + (uint64_t)chunk * (uint64_t)(kChunkFloats * 4);

    // D# group 0 (128b): count=1 valid user descriptor, lds_addr, global
    // tile address, type=2 ("image").
    u32x4 g0;
    g0[0] = 1u;                                        // count=1, is_restore=0, gather=0
    g0[1] = lds_base;                                  // lds_addr (bytes)
    g0[2] = (uint32_t)gaddr;                           // global_addr[31:0]
    g0[3] = (uint32_t)(gaddr >> 32) | (2u << 30);      // global_addr[56:32] | type=2

    // D# group 1 (256b): data_size=4B, 1-D tile of 16384 elements.
    u32x8 g1;
    g1[0] = 2u << 16;                                  // data_size=2 -> 4 bytes
    g1[1] = ((uint32_t)kChunkFloats & 0xFFFFu) << 16;  // tensor_dim0[15:0] @ bits 63:48
    g1[2] = ((uint32_t)kChunkFloats >> 16) | (1u << 16); // tensor_dim0[31:16], tensor_dim1=1
    g1[3] = (uint32_t)kChunkFloats << 16;              // tile_dim0 = 16384 @ bits 127:112
    g1[4] = 0u;                                        // tile_dim1=0, tile_dim2=0 (1-D)
    g1[5] = (uint32_t)kChunkFloats;                    // tensor_dim0_stride[31:0]
    g1[6] = 0u;                                        // stride msbs / dim1_stride lo
    g1[7] = 0u;                                        // dim1_stride hi
    asm volatile("tensor_load_to_lds %0, %1" : : "s"(g0), "s"(g1) : "memory");
    __builtin_amdgcn_s_wait_tensorcnt(0);
  }
  __syncthreads();  // LDS tile now valid for all 8 waves

  float* __restrict__ obase = out + (uint64_t)chunk * (uint64_t)kChunkFloats;

  // Each thread handles 4 patches; consecutive lanes -> consecutive 16B
  // segments (512 B contiguous per wave per row) for coalesced NT stores.
#pragma unroll
  for (int j = 0; j < kPatchesPerChunk / 256; ++j) {
    const uint32_t p   = threadIdx.x + (uint32_t)j * 256u;       // patch id in chunk
    const uint32_t off = (p >> 6) * (4u * kW) + (p & 63u) * 4u;  // float offset, row 0

    const v4f r0 = *(const v4f*)(tile + off);
    const v4f r1 = *(const v4f*)(tile + off + kW);
    const v4f r2 = *(const v4f*)(tile + off + 2 * kW);
    const v4f r3 = *(const v4f*)(tile + off + 3 * kW);

    const float m = fmaxf(hmax4(vmax4(r0, r1)), hmax4(vmax4(r2, r3)));

    const v4f e0 = vexp4(r0, m);
    const v4f e1 = vexp4(r1, m);
    const v4f e2 = vexp4(r2, m);
    const v4f e3 = vexp4(r3, m);

    const float inv = 1.0f / ((hsum4(e0) + hsum4(e1)) + (hsum4(e2) + hsum4(e3)));

    __builtin_nontemporal_store(e0 * inv, (v4f*)(obase + off));
    __builtin_nontemporal_store(e1 * inv, (v4f*)(obase + off + kW));
    __builtin_nontemporal_store(e2 * inv, (v4f*)(obase + off + 2 * kW));
    __builtin_nontemporal_store(e3 * inv, (v4f*)(obase + off + 3 * kW));
  }
}

extern "C" void kernel_launch(void* const* d_in, const int* in_sizes, int n_in,
                              void* d_out, int out_size, void* d_ws, size_t ws_size,
                              hipStream_t stream) {
  (void)n_in; (void)out_size; (void)d_ws; (void)ws_size;
  const float* x   = (const float*)d_in[0];
  float*       out = (float*)d_out;
  // 16*64*256*256 elements / 16384 per chunk = 4096 workgroups
  const int nChunks = in_sizes[0] / kChunkFloats;
  patch_softmax_tdm<<<dim3(nChunks), dim3(256), 0, stream>>>(x, out);
}